// GINModule_82317343195433
// MI455X (gfx1250) — compile-verified
//
#include <hip/hip_runtime.h>

typedef __attribute__((ext_vector_type(2))) float v2f;
typedef __attribute__((ext_vector_type(8))) float v8f;

#define NPTS   65536
#define DIM    64
#define HID    128
#define NBATCH 16
#define KNN    16
#define NPB    (NPTS / NBATCH)   // 4096 points per batch
#define CTILE  256               // candidate tile staged in LDS

#define WAVES_PER_BLOCK 4
#define ROWS_PER_WAVE   16
#define H0S (DIM + 4)            // padded LDS stride for h0
#define H1S (HID + 4)            // padded LDS stride for h1

// ---------------------------------------------------------------------------
// Kernel 1: exact KNN (top-16 smallest squared distances) per point, within
// its batch. One thread per query point; candidates staged via LDS tiles.
// ---------------------------------------------------------------------------
__global__ __launch_bounds__(CTILE) void knn_kernel(
    const float* __restrict__ pos, int* __restrict__ knn_idx)
{
    __shared__ float sx[CTILE], sy[CTILE], sz[CTILE];

    const int q    = blockIdx.x * blockDim.x + threadIdx.x;
    const int b    = q / NPB;            // matches batch = arange // (N//B)
    const int base = b * NPB;

    const float px = pos[q * 3 + 0];
    const float py = pos[q * 3 + 1];
    const float pz = pos[q * 3 + 2];

    float dist_k[KNN];
    int   idx_k[KNN];
#pragma unroll
    for (int i = 0; i < KNN; ++i) { dist_k[i] = __builtin_inff(); idx_k[i] = q; }
    float cur_max  = __builtin_inff();
    int   max_slot = 0;

    for (int t = 0; t < NPB; t += CTILE) {
        __syncthreads();
        {
            const int c = base + t + threadIdx.x;
            sx[threadIdx.x] = pos[c * 3 + 0];
            sy[threadIdx.x] = pos[c * 3 + 1];
            sz[threadIdx.x] = pos[c * 3 + 2];
        }
        // Speculative prefetch of the next candidate tile (global_prefetch_b8).
        if (t + CTILE < NPB) {
            const int cn = base + t + CTILE + threadIdx.x;
            __builtin_prefetch(pos + (size_t)cn * 3, 0, 3);
        }
        __syncthreads();

        for (int j = 0; j < CTILE; ++j) {
            const int   cj = base + t + j;
            const float dx = px - sx[j];
            const float dy = py - sy[j];
            const float dz = pz - sz[j];
            const float d2 = dx * dx + dy * dy + dz * dz;
            if (cj == q) continue;                 // exclude self (inf diagonal)
            if (d2 < cur_max) {
                // Replace current worst; keeps exact top-16 set.
#pragma unroll
                for (int i = 0; i < KNN; ++i)
                    if (i == max_slot) { dist_k[i] = d2; idx_k[i] = cj; }
                // Lazy rescan for new worst.
                cur_max = dist_k[0]; max_slot = 0;
#pragma unroll
                for (int i = 1; i < KNN; ++i)
                    if (dist_k[i] > cur_max) { cur_max = dist_k[i]; max_slot = i; }
            }
        }
    }

#pragma unroll
    for (int i = 0; i < KNN; ++i) knn_idx[q * KNN + i] = idx_k[i];
}

// ---------------------------------------------------------------------------
// Kernel 2: fused neighbor-sum + MLP (relu(h0@W1+b1))@W2+b2 using
// V_WMMA_F32_16X16X4_F32. One wave per 16-row tile.
//
// ISA fragment layouts (CDNA5 ISA):
//   A (16x4 f32):  lane l: m = l&15, K = (l>>4)*2 + {0,1}  -> v2f
//   B (4x16 f32):  lane l: n = l&15, K = (l>>4)*2 + {0,1}  -> v2f (symmetric)
//   C/D (16x16):   lane l, vgpr i: m = (l>>4)*8 + i, n = l&15 -> v8f
// ---------------------------------------------------------------------------
__global__ __launch_bounds__(WAVES_PER_BLOCK * 32) void gnn_mlp_kernel(
    const float* __restrict__ x, const int* __restrict__ knn_idx,
    const float* __restrict__ W1, const float* __restrict__ b1,
    const float* __restrict__ W2, const float* __restrict__ b2,
    float* __restrict__ out)
{
    __shared__ float h0[WAVES_PER_BLOCK][ROWS_PER_WAVE][H0S];
    __shared__ float h1[WAVES_PER_BLOCK][ROWS_PER_WAVE][H1S];

    const int lane     = threadIdx.x & 31;
    const int wave     = threadIdx.x >> 5;
    const int tile_row = (blockIdx.x * WAVES_PER_BLOCK + wave) * ROWS_PER_WAVE;
    const int half     = lane >> 4;   // 0 or 1: selects K-pair / M-half
    const int mn       = lane & 15;   // A-row / B-col / C-col index

    // ---- Phase 1: h0 = x[row] + sum_k x[nbr_k]  (each lane owns 2 columns)
    {
        const int c0 = lane * 2;
        for (int r = 0; r < ROWS_PER_WAVE; ++r) {
            const int row = tile_row + r;
            v2f acc = *(const v2f*)(x + (size_t)row * DIM + c0);
#pragma unroll 4
            for (int k = 0; k < KNN; ++k) {
                const int nb = knn_idx[row * KNN + k];
                acc += *(const v2f*)(x + (size_t)nb * DIM + c0);
            }
            h0[wave][r][c0]     = acc.x;
            h0[wave][r][c0 + 1] = acc.y;
        }
    }
    __syncthreads();

    // Preload all GEMM1 A-fragments (K=64 -> 16 k-steps of 4).
    v2f afrag[DIM / 4];
#pragma unroll
    for (int kt = 0; kt < DIM / 4; ++kt) {
        const int kb = kt * 4 + half * 2;
        afrag[kt].x = h0[wave][mn][kb];
        afrag[kt].y = h0[wave][mn][kb + 1];
    }

    // ---- Phase 2: h1 = relu(h0 @ W1 + b1), 16x128 tile
    for (int nt = 0; nt < HID / 16; ++nt) {
        const int n = nt * 16 + mn;
        const float bias = b1[n];
        v8f c;
#pragma unroll
        for (int i = 0; i < 8; ++i) c[i] = bias;
#pragma unroll
        for (int kt = 0; kt < DIM / 4; ++kt) {
            const int kb = kt * 4 + half * 2;
            v2f bfrag;
            bfrag.x = W1[kb * HID + n];
            bfrag.y = W1[(kb + 1) * HID + n];
            c = __builtin_amdgcn_wmma_f32_16x16x4_f32(
                    false, afrag[kt], false, bfrag, (short)0, c, false, false);
        }
#pragma unroll
        for (int i = 0; i < 8; ++i) {
            const float v = c[i] > 0.0f ? c[i] : 0.0f;   // ReLU
            h1[wave][half * 8 + i][n] = v;
        }
    }
    __syncthreads();

    // ---- Phase 3: out = h1 @ W2 + b2, 16x64 tile
    for (int nt = 0; nt < DIM / 16; ++nt) {
        const int n = nt * 16 + mn;
        const float bias = b2[n];
        v8f c;
#pragma unroll
        for (int i = 0; i < 8; ++i) c[i] = bias;
#pragma unroll
        for (int kt = 0; kt < HID / 4; ++kt) {
            const int kb = kt * 4 + half * 2;
            v2f a;
            a.x = h1[wave][mn][kb];
            a.y = h1[wave][mn][kb + 1];
            v2f bfrag;
            bfrag.x = W2[kb * DIM + n];
            bfrag.y = W2[(kb + 1) * DIM + n];
            c = __builtin_amdgcn_wmma_f32_16x16x4_f32(
                    false, a, false, bfrag, (short)0, c, false, false);
        }
#pragma unroll
        for (int i = 0; i < 8; ++i) {
            const int m = half * 8 + i;
            out[(size_t)(tile_row + m) * DIM + n] = c[i];
        }
    }
}

// ---------------------------------------------------------------------------
extern "C" void kernel_launch(void* const* d_in, const int* in_sizes, int n_in,
                              void* d_out, int out_size, void* d_ws, size_t ws_size,
                              hipStream_t stream) {
    const float* x   = (const float*)d_in[0];
    const float* pos = (const float*)d_in[1];
    // d_in[2] = batch; recomputed as q / (N/B) on device.
    const float* W1  = (const float*)d_in[3];
    const float* b1  = (const float*)d_in[4];
    const float* W2  = (const float*)d_in[5];
    const float* b2  = (const float*)d_in[6];
    float* out   = (float*)d_out;
    int* knn_idx = (int*)d_ws;   // N*K ints = 4 MB scratch

    knn_kernel<<<NPTS / CTILE, CTILE, 0, stream>>>(pos, knn_idx);

    const int tiles  = NPTS / ROWS_PER_WAVE;             // 4096 wave-tiles
    const int blocks = tiles / WAVES_PER_BLOCK;          // 1024 blocks
    gnn_mlp_kernel<<<blocks, WAVES_PER_BLOCK * 32, 0, stream>>>(
        x, knn_idx, W1, b1, W2, b2, out);
}